// FAGCN_2465311228223
// MI455X (gfx1250) — compile-verified
//
#include <hip/hip_runtime.h>
#include <hip/hip_bf16.h>
#include <math.h>

#define EPSV 0.3f
#define FDIM 512
#define HDIM 128
#define CDIM 40
#define LDSS 516   // 512 + 4 pad -> lanes 0..15 hit distinct LDS banks (4*m mod 64 unique)

typedef __attribute__((ext_vector_type(2))) float v2f;
typedef __attribute__((ext_vector_type(8))) float v8f;

// ---------------- degree / norm ----------------
__global__ void k_deg(const int* __restrict__ row, float* __restrict__ deg, int E) {
  int e = blockIdx.x * blockDim.x + threadIdx.x;
  if (e < E) atomicAdd(&deg[row[e]], 1.0f);
}

__global__ void k_nd(float* __restrict__ d, int N) {
  int n = blockIdx.x * blockDim.x + threadIdx.x;
  if (n < N) d[n] = 1.0f / sqrtf(fmaxf(d[n], 1.0f));
}

// ---------------- GEMM1: h = relu(x @ t1_w^T + b), via V_WMMA_F32_16X16X4_F32 ----------------
// Block: 256 threads = 8 waves. Block computes 32 rows x all 128 cols.
// Wave w owns cols [16w, 16w+16) with TWO accumulators (row halves m0..m0+15 and
// m0+16..m0+31) so each B fragment (t1_w) is reused twice -> halves L2 traffic on t1_w.
// Tail block (N % 32 == 16): LDS loads clamp the row index, stores are guarded; WMMA
// itself runs with EXEC all-ones as required.
__global__ void __launch_bounds__(256) k_gemm1(const float* __restrict__ x,
                                               const float* __restrict__ w,
                                               const float* __restrict__ bias,
                                               float* __restrict__ h, int N) {
  __shared__ float xs[32 * LDSS];
  const int m0 = blockIdx.x * 32;
  const int tid = threadIdx.x;
  // cooperative coalesced load of the 32x512 x-tile (row clamped for the tail block)
  for (int i = tid; i < 32 * FDIM; i += 256) {
    int r = i >> 9;            // / 512
    int c = i & (FDIM - 1);
    int gr = m0 + r;
    if (gr >= N) gr = N - 1;
    xs[r * LDSS + c] = x[(size_t)gr * FDIM + c];
  }
  __syncthreads();

  const int wave = tid >> 5;
  const int lane = tid & 31;
  const int half = lane >> 4;   // 0: K pair {0,1}; 1: K pair {2,3}
  const int lm   = lane & 15;
  const int n0   = wave * 16;   // 8 waves cover H = 128

  const float* wrow  = w + (size_t)(n0 + lm) * FDIM;   // t1_w row = B column n0+lm
  const float* arow0 = xs + lm * LDSS;                 // A rows M = lm      (half 0)
  const float* arow1 = xs + (16 + lm) * LDSS;          // A rows M = 16+lm   (half 1)

  v8f acc0 = {};
  v8f acc1 = {};
  for (int k = 0; k < FDIM; k += 4) {
    int kk = k + 2 * half;
    v2f b;  b.x  = wrow[kk];  b.y  = wrow[kk + 1];     // B 4x16 layout
    v2f a0; a0.x = arow0[kk]; a0.y = arow0[kk + 1];    // A 16x4 layout (ISA 7.12.2)
    v2f a1; a1.x = arow1[kk]; a1.y = arow1[kk + 1];
    acc0 = __builtin_amdgcn_wmma_f32_16x16x4_f32(false, a0, false, b,
                                                 (short)0, acc0, false, false);
    acc1 = __builtin_amdgcn_wmma_f32_16x16x4_f32(false, a1, false, b,
                                                 (short)0, acc1, false, false);
  }

  // C/D layout: VGPR v -> M = v + 8*half, N = lm
  float bc = bias[n0 + lm];
  #pragma unroll
  for (int v = 0; v < 8; v++) {
    int m = m0 + v + 8 * half;
    if (m < N)
      h[(size_t)m * HDIM + n0 + lm] = fmaxf(acc0[v] + bc, 0.0f);
  }
  #pragma unroll
  for (int v = 0; v < 8; v++) {
    int m = m0 + 16 + v + 8 * half;
    if (m < N)
      h[(size_t)m * HDIM + n0 + lm] = fmaxf(acc1[v] + bc, 0.0f);
  }
}

// ---------------- per-node gate projections: s1 = <h[n], gw1>, s2 = <h[n], gw2> ----------------
__global__ void k_gate(const float* __restrict__ h, const float* __restrict__ g1,
                       const float* __restrict__ g2, float* __restrict__ s1,
                       float* __restrict__ s2, int N) {
  int gi = blockIdx.x * blockDim.x + threadIdx.x;
  int wv = gi >> 5, lane = gi & 31;
  if (wv >= N) return;
  const float* hn = h + (size_t)wv * HDIM;
  float a = 0.0f, b = 0.0f;
  #pragma unroll
  for (int j = lane; j < HDIM; j += 32) {
    float v = hn[j];
    a += v * g1[j];
    b += v * g2[j];
  }
  #pragma unroll
  for (int off = 16; off; off >>= 1) {
    a += __shfl_xor(a, off);
    b += __shfl_xor(b, off);
  }
  if (lane == 0) { s1[wv] = a; s2[wv] = b; }
}

// ---------------- edge aggregation: agg[col] += tanh(s1[row]+s2[col]+gb)*nd_r*nd_c * h[row] ----
__global__ void k_edge(const float* __restrict__ h, const float* __restrict__ s1,
                       const float* __restrict__ s2, const float* __restrict__ nd,
                       const int* __restrict__ row, const int* __restrict__ col,
                       const float* __restrict__ gb, int l,
                       float* __restrict__ agg, int E) {
  int gi = blockIdx.x * blockDim.x + threadIdx.x;
  int e = gi >> 5, lane = gi & 31;
  if (e >= E) return;                       // uniform per wave
  int r = row[e], c = col[e];
  float g = tanhf(s1[r] + s2[c] + gb[l]);
  float wgt = g * nd[r] * nd[c];
  const float* hr = h + (size_t)r * HDIM;
  float* ac = agg + (size_t)c * HDIM;
  #pragma unroll
  for (int j = lane; j < HDIM; j += 32)
    atomicAdd(&ac[j], wgt * hr[j]);
}

// ---------------- h_next = EPS*raw + agg (in place on agg) ----------------
__global__ void k_combine(const float* __restrict__ raw, float* __restrict__ agg, int total) {
  int i = blockIdx.x * blockDim.x + threadIdx.x;
  if (i < total) agg[i] = EPSV * raw[i] + agg[i];
}

// ---------------- GEMM2: logits = h @ t2_w^T + b ----------------
__global__ void k_gemm2(const float* __restrict__ h, const float* __restrict__ w,
                        const float* __restrict__ b, float* __restrict__ out, int N) {
  int idx = blockIdx.x * blockDim.x + threadIdx.x;
  if (idx >= N * CDIM) return;
  int n = idx / CDIM, c = idx % CDIM;
  const float4* hn = (const float4*)(h + (size_t)n * HDIM);
  const float4* wc = (const float4*)(w + (size_t)c * HDIM);
  float acc = b[c];
  #pragma unroll
  for (int k = 0; k < HDIM / 4; k++) {
    float4 hv = hn[k], wv = wc[k];
    acc += hv.x * wv.x + hv.y * wv.y + hv.z * wv.z + hv.w * wv.w;
  }
  out[idx] = acc;
}

// ---------------- in-place log_softmax over 40 classes, one wave per node ----------------
__global__ void k_lsm(float* __restrict__ out, int N) {
  int gi = blockIdx.x * blockDim.x + threadIdx.x;
  int wv = gi >> 5, lane = gi & 31;
  if (wv >= N) return;
  float* o = out + (size_t)wv * CDIM;
  float v0 = (lane < CDIM) ? o[lane] : -INFINITY;
  float v1 = (lane + 32 < CDIM) ? o[lane + 32] : -INFINITY;
  float m = fmaxf(v0, v1);
  #pragma unroll
  for (int off = 16; off; off >>= 1) m = fmaxf(m, __shfl_xor(m, off));
  float s = ((lane < CDIM) ? expf(v0 - m) : 0.0f) +
            ((lane + 32 < CDIM) ? expf(v1 - m) : 0.0f);
  #pragma unroll
  for (int off = 16; off; off >>= 1) s += __shfl_xor(s, off);
  float ls = m + logf(s);
  if (lane < CDIM)       o[lane]       = v0 - ls;
  if (lane + 32 < CDIM)  o[lane + 32]  = v1 - ls;
}

extern "C" void kernel_launch(void* const* d_in, const int* in_sizes, int n_in,
                              void* d_out, int out_size, void* d_ws, size_t ws_size,
                              hipStream_t stream) {
  const float* x      = (const float*)d_in[0];
  const float* t1_w   = (const float*)d_in[1];
  const float* t1_b   = (const float*)d_in[2];
  const float* t2_w   = (const float*)d_in[3];
  const float* t2_b   = (const float*)d_in[4];
  const float* gate_w = (const float*)d_in[5];
  const float* gate_b = (const float*)d_in[6];
  const int*   ei     = (const int*)d_in[7];

  const int N = in_sizes[0] / FDIM;   // 50000
  const int E = in_sizes[7] / 2;      // 800000
  const int* row = ei;
  const int* col = ei + E;

  // workspace carve-up
  char* ws = (char*)d_ws;
  size_t off = 0;
  auto carve = [&](size_t bytes) -> float* {
    float* p = (float*)(ws + off);
    off = (off + bytes + 255) & ~(size_t)255;
    return p;
  };
  const size_t nhBytes = (size_t)N * HDIM * sizeof(float);
  float* raw = carve(nhBytes);        // relu(x W1^T + b) — kept for EPS*raw
  float* b1  = carve(nhBytes);        // layer-1 agg -> h1 (in place)
  float* b2  = carve(nhBytes);        // layer-2 agg -> h2 (in place)
  float* s1  = carve((size_t)N * sizeof(float));
  float* s2  = carve((size_t)N * sizeof(float));
  float* nd  = carve((size_t)N * sizeof(float));   // deg, then nd in place

  float* out = (float*)d_out;

  // 1) degree + norm
  hipMemsetAsync(nd, 0, (size_t)N * sizeof(float), stream);
  k_deg<<<(E + 255) / 256, 256, 0, stream>>>(row, nd, E);
  k_nd<<<(N + 255) / 256, 256, 0, stream>>>(nd, N);

  // 2) GEMM1 with WMMA, 32 rows per block (tail block clamps/guards)
  k_gemm1<<<(N + 31) / 32, 256, 0, stream>>>(x, t1_w, t1_b, raw, N);

  // 3) two FAGCN layers
  const float* hcur = raw;
  float* aggbuf[2] = { b1, b2 };
  for (int l = 0; l < 2; l++) {
    const float* g1 = gate_w + (size_t)l * 2 * HDIM;
    const float* g2 = g1 + HDIM;
    k_gate<<<(N * 32 + 255) / 256, 256, 0, stream>>>(hcur, g1, g2, s1, s2, N);
    hipMemsetAsync(aggbuf[l], 0, nhBytes, stream);
    k_edge<<<((size_t)E * 32 + 255) / 256, 256, 0, stream>>>(hcur, s1, s2, nd, row, col,
                                                             gate_b, l, aggbuf[l], E);
    k_combine<<<(N * HDIM + 255) / 256, 256, 0, stream>>>(raw, aggbuf[l], N * HDIM);
    hcur = aggbuf[l];
  }

  // 4) classifier + log_softmax (in place on d_out)
  k_gemm2<<<(N * CDIM + 255) / 256, 256, 0, stream>>>(hcur, t2_w, t2_b, out, N);
  k_lsm<<<(N * 32 + 255) / 256, 256, 0, stream>>>(out, N);
}